// Ewald_9698036154711
// MI455X (gfx1250) — compile-verified
//
#include <hip/hip_runtime.h>
#include <hip/hip_bf16.h>
#include <math.h>

// CDNA5 / gfx1250 wave32 WMMA types
typedef __attribute__((ext_vector_type(2))) float v2f;
typedef __attribute__((ext_vector_type(4))) float v4f;
typedef __attribute__((ext_vector_type(8))) float v8f;

#define N_ATOMS 4096
#define NQ 8
#define TILE 16
#define TILES (N_ATOMS / TILE)          // 256
#define WAVES_PER_BLOCK 8               // 256 threads = 8 wave32
#define THREADS (WAVES_PER_BLOCK * 32)
#define NBLOCKS (TILES * TILES / WAVES_PER_BLOCK)   // 8192

// compile-time literals (avoid s_load of __constant__ memory)
#define INV_SQRT2   0.70710678118654752440f
// NORM_FACTOR / (2*pi) / 2
#define FINAL_SCALE (90.0474f / (4.0f * 3.14159265358979323846f))

// Branchless erf for x >= 0 (Abramowitz & Stegun 7.1.26, max abs err ~1.5e-7).
__device__ __forceinline__ float erf_pos(float x) {
    const float t = __builtin_amdgcn_rcpf(fmaf(0.3275911f, x, 1.0f));
    float p = fmaf(1.061405429f, t, -1.453152027f);
    p = fmaf(p, t, 1.421413741f);
    p = fmaf(p, t, -0.284496736f);
    p = fmaf(p, t, 0.254829592f);
    p = p * t;
    return fmaf(-p, __expf(-x * x), 1.0f);
}

// One wave per 16x16 tile of the (i,j) pair matrix, upper triangle only
// (tj >= ti), off-diagonal tiles doubled by symmetry.
__global__ __launch_bounds__(THREADS)
void Ewald_tile_kernel(const float* __restrict__ q,   // [4096, 8]
                       const float* __restrict__ r,   // [4096, 3]
                       float* __restrict__ block_sums)
{
    const int lane  = threadIdx.x & 31;
    const int wave  = threadIdx.x >> 5;
    const int wid   = blockIdx.x * WAVES_PER_BLOCK + wave;  // 0..65535
    const int ti    = wid >> 8;      // tile row   0..255
    const int tj    = wid & 255;     // tile col   0..255

    float lsum = 0.0f;

    if (tj >= ti) {
        const int half = lane >> 4;   // 0: K=0..1 / M=0..7   1: K=2..3 / M=8..15
        const int lrow = lane & 15;

        // ---- Gram tile G = Qi (16x8) * Qj^T (8x16) via two f32 WMMAs (K=4 each) ----
        // A 16x4 f32 layout: lane(l<16) row=l holds K={0,1}; lanes 16..31 hold K={2,3}.
        // B 4x16 f32 layout mirrors it -> identical per-lane load pattern on Qj.
        const float* qi = q + (ti * TILE + lrow) * NQ + 2 * half;
        const float* qj = q + (tj * TILE + lrow) * NQ + 2 * half;
        v2f a0 = *(const v2f*)(qi);        // channels kbase=0
        v2f b0 = *(const v2f*)(qj);
        v2f a1 = *(const v2f*)(qi + 4);    // channels kbase=4
        v2f b1 = *(const v2f*)(qj + 4);

        v8f c = {};
        c = __builtin_amdgcn_wmma_f32_16x16x4_f32(false, a0, false, b0,
                                                  (short)0, c, false, false);
        c = __builtin_amdgcn_wmma_f32_16x16x4_f32(false, a1, false, b1,
                                                  (short)0, c, false, false);

        // ---- preload this lane's 8 r_i rows: 96 contiguous, 16B-aligned bytes ----
        // C/D layout: VGPR v, lane l -> M = v + 8*(l>=16), N = l%16.
        const int i0 = ti * TILE + 8 * half;       // multiple of 8 -> 3*i0*4 % 16 == 0
        float rbuf[24];
        {
            const v4f* rp = (const v4f*)(r + 3 * i0);
            #pragma unroll
            for (int t = 0; t < 6; ++t) ((v4f*)rbuf)[t] = rp[t];
        }

        const int j = tj * TILE + lrow;
        const float rjx = r[3 * j + 0];
        const float rjy = r[3 * j + 1];
        const float rjz = r[3 * j + 2];

        // ---- branchless elementwise K_ij = erf(d/sqrt2)/d on lane's 8 pairs ----
        #pragma unroll
        for (int v = 0; v < 8; ++v) {
            const float dx = rbuf[3 * v + 0] - rjx;
            const float dy = rbuf[3 * v + 1] - rjy;
            const float dz = rbuf[3 * v + 2] - rjz;
            float d2 = fmaf(dx, dx, fmaf(dy, dy, dz * dz));
            d2 = fmaxf(d2, 1e-12f);                       // diagonal guard (masked below)
            const float rs = __builtin_amdgcn_rsqf(d2);   // 1/d
            const float d  = d2 * rs;                     // d
            const float kv = erf_pos(d * INV_SQRT2) * rs; // erf(d/sqrt2)/d
            const float contrib = kv * c[v];
            lsum += (i0 + v != j) ? contrib : 0.0f;       // cndmask, no divergence
        }
        if (ti != tj) lsum *= 2.0f;  // symmetry: upper triangle counted twice
    }

    // ---- wave32 reduction ----
    #pragma unroll
    for (int off = 16; off > 0; off >>= 1)
        lsum += __shfl_xor(lsum, off, 32);

    __shared__ float wsum[WAVES_PER_BLOCK];
    if (lane == 0) wsum[wave] = lsum;
    __syncthreads();
    if (threadIdx.x == 0) {
        float s = 0.0f;
        #pragma unroll
        for (int w = 0; w < WAVES_PER_BLOCK; ++w) s += wsum[w];
        block_sums[blockIdx.x] = s;
    }
}

// Deterministic single-block final reduction (no float atomics -> bitwise stable).
__global__ __launch_bounds__(256)
void Ewald_reduce_kernel(const float* __restrict__ block_sums,
                         float* __restrict__ out, int n)
{
    __shared__ float sm[256];
    float s = 0.0f;
    for (int idx = threadIdx.x; idx < n; idx += 256) s += block_sums[idx];
    sm[threadIdx.x] = s;
    __syncthreads();
    for (int off = 128; off > 0; off >>= 1) {
        if ((int)threadIdx.x < off) sm[threadIdx.x] += sm[threadIdx.x + off];
        __syncthreads();
    }
    if (threadIdx.x == 0) out[0] = sm[0] * FINAL_SCALE;
}

extern "C" void kernel_launch(void* const* d_in, const int* in_sizes, int n_in,
                              void* d_out, int out_size, void* d_ws, size_t ws_size,
                              hipStream_t stream) {
    const float* q = (const float*)d_in[0];   // [4096, 8]  float32
    const float* r = (const float*)d_in[1];   // [4096, 3]  float32
    // d_in[2] = cell, singular -> real-space branch; unused.
    float* out = (float*)d_out;               // [1] float32
    float* block_sums = (float*)d_ws;         // NBLOCKS floats (32 KB)

    Ewald_tile_kernel<<<NBLOCKS, THREADS, 0, stream>>>(q, r, block_sums);
    Ewald_reduce_kernel<<<1, 256, 0, stream>>>(block_sums, out, NBLOCKS);
}